// GATConvMultiQuant_49194555408767
// MI455X (gfx1250) — compile-verified
//
#include <hip/hip_runtime.h>
#include <hip/hip_bf16.h>
#include <math.h>

// ---------------- problem constants ----------------
#define NNODES 50000
#define NEDGES 800000
#define ETOT   (NEDGES + NNODES)   // edges + self loops = 850000
#define INFEAT 128
#define HEADS  4
#define CH     32
#define OUTF   128                 // HEADS*CH
#define LVL_LO 15.0f               // 2^4 - 1
#define LVL_HI 255.0f              // 2^8 - 1
#define NSLOPE 0.2f

// scalar reduction slot indices (encoded uint32 min/max; even=min, odd=max)
enum {
  SL_X   = 0,    // x:    hi_mn, hi_mx, lo_mn, lo_mx
  SL_W   = 4,    // weight: mn, mx   (paired with att in one commit)
  SL_ATT = 6,    // att:    mn, mx
  SL_F   = 8,    // feat (x_q @ w_q): hi/lo mn/mx
  SL_A   = 12,   // alpha pre-softmax
  SL_M   = 16,   // msg
  SL_G   = 20,   // aggr
  SL_O   = 24,   // out (aggr_q + bias)
  SL_COUNT = 28
};

typedef float v2f __attribute__((ext_vector_type(2)));
typedef float v8f __attribute__((ext_vector_type(8)));

// ---------------- helpers ----------------
// monotone float <-> uint encoding so min/max reductions can use integer atomics
__device__ __forceinline__ unsigned fenc(float f) {
  unsigned u = __float_as_uint(f);
  return (u & 0x80000000u) ? ~u : (u | 0x80000000u);
}
__device__ __forceinline__ float fdec(unsigned u) {
  unsigned v = (u & 0x80000000u) ? (u & 0x7fffffffu) : ~u;
  return __uint_as_float(v);
}
__device__ __forceinline__ float finite0(float v) { return isfinite(v) ? v : 0.0f; }

// fake-quant: round((x-mn)/scale)*scale + mn, scale = max((mx-mn)/levels, 1e-8)
// rintf = round-half-even, matching jnp.round
__device__ __forceinline__ float fq_apply(float x, float mn, float mx, float levels) {
  float scale = fmaxf((mx - mn) / levels, 1e-8f);
  return rintf((x - mn) / scale) * scale + mn;
}

// block-level commit of two (min,max) pairs into global encoded slots
__device__ __forceinline__ void commit_minmax4(unsigned* slots, int base,
                                               float mnA, float mxA,
                                               float mnB, float mxB,
                                               unsigned* sred) {
  if (threadIdx.x == 0) {
    sred[0] = 0xFFFFFFFFu; sred[1] = 0u; sred[2] = 0xFFFFFFFFu; sred[3] = 0u;
  }
  __syncthreads();
  atomicMin(&sred[0], fenc(mnA));
  atomicMax(&sred[1], fenc(mxA));
  atomicMin(&sred[2], fenc(mnB));
  atomicMax(&sred[3], fenc(mxB));
  __syncthreads();
  if (threadIdx.x == 0) {
    atomicMin(&slots[base + 0], sred[0]);
    atomicMax(&slots[base + 1], sred[1]);
    atomicMin(&slots[base + 2], sred[2]);
    atomicMax(&slots[base + 3], sred[3]);
  }
}

#define GRID_STRIDE(i, total) \
  for (long long i = blockIdx.x * (long long)blockDim.x + threadIdx.x; \
       i < (total); i += (long long)gridDim.x * blockDim.x)

// ---------------- K0: init outputs / slots / segment state ----------------
__global__ __launch_bounds__(256) void k_init(float* out, unsigned* slots,
                                              unsigned* nodeMax, float* nodeSum) {
  GRID_STRIDE(i, (long long)NNODES * OUTF) {
    out[i] = 0.0f;
    if (i < SL_COUNT) slots[i] = (i & 1) ? 0u : 0xFFFFFFFFu;
    if (i < (long long)NNODES * HEADS) { nodeMax[i] = 0u; nodeSum[i] = 0.0f; }
  }
}

// ---------------- K1: masked min/max of x ----------------
__global__ __launch_bounds__(256) void k_minmax_x(const float* x, const unsigned char* mask,
                                                  unsigned* slots) {
  __shared__ unsigned s4[4];
  float mnH = INFINITY, mxH = -INFINITY, mnL = INFINITY, mxL = -INFINITY;
  GRID_STRIDE(i, (long long)NNODES * INFEAT) {
    float v = x[i];
    if (mask[i >> 7]) { mnH = fminf(mnH, v); mxH = fmaxf(mxH, v); }
    else              { mnL = fminf(mnL, v); mxL = fmaxf(mxL, v); }
  }
  commit_minmax4(slots, SL_X, mnH, mxH, mnL, mxL, s4);
}

// ---------------- K2: min/max of weight + att (single block) ----------------
__global__ __launch_bounds__(256) void k_minmax_watt(const float* w, const float* att,
                                                     unsigned* slots) {
  __shared__ unsigned s4[4];
  float wmn = INFINITY, wmx = -INFINITY, amn = INFINITY, amx = -INFINITY;
  for (int i = threadIdx.x; i < INFEAT * OUTF; i += blockDim.x) {
    float v = w[i]; wmn = fminf(wmn, v); wmx = fmaxf(wmx, v);
  }
  for (int i = threadIdx.x; i < HEADS * 2 * CH; i += blockDim.x) {
    float v = att[i]; amn = fminf(amn, v); amx = fmaxf(amx, v);
  }
  commit_minmax4(slots, SL_W, wmn, wmx, amn, amx, s4);  // SL_W..SL_ATT+1 contiguous
}

// ---------------- K2b: quantize weight (4-bit fq) and att ----------------
__global__ __launch_bounds__(256) void k_quant_watt(const float* w, const float* att,
                                                    const unsigned* slots,
                                                    float* w_q, float* att_q) {
  float wmn = finite0(fdec(slots[SL_W]));   float wmx = finite0(fdec(slots[SL_W + 1]));
  float amn = finite0(fdec(slots[SL_ATT])); float amx = finite0(fdec(slots[SL_ATT + 1]));
  int i = blockIdx.x * blockDim.x + threadIdx.x;
  if (i < INFEAT * OUTF)   w_q[i]   = fq_apply(w[i],   wmn, wmx, LVL_LO);
  if (i < HEADS * 2 * CH)  att_q[i] = fq_apply(att[i], amn, amx, LVL_LO);
}

// ---------------- K3: feat = x_q @ w_q via V_WMMA_F32_16X16X4_F32 ----------
// block = 256 threads (8 waves); block owns 16 rows of x, each wave owns one
// 16-wide N tile -> full 16x128 output tile per block. x tile is quantized on
// the fly into LDS (stride 132 words => conflict-free A reads).
#define AS_STRIDE 132
__global__ __launch_bounds__(256) void k_gemm_feat(const float* x, const unsigned char* mask,
                                                   unsigned* slots, const float* w_q,
                                                   float* feat) {
  __shared__ float As[16 * AS_STRIDE];
  __shared__ unsigned s4[4];
  float xmnH = finite0(fdec(slots[SL_X + 0])), xmxH = finite0(fdec(slots[SL_X + 1]));
  float xmnL = finite0(fdec(slots[SL_X + 2])), xmxL = finite0(fdec(slots[SL_X + 3]));

  int m0 = blockIdx.x * 16;
  for (int e = threadIdx.x; e < 16 * INFEAT; e += blockDim.x) {
    int r = e >> 7, c = e & 127;
    int row = m0 + r;
    float v = x[(long long)row * INFEAT + c];
    v = mask[row] ? fq_apply(v, xmnH, xmxH, LVL_HI)
                  : fq_apply(v, xmnL, xmxL, LVL_LO);
    As[r * AS_STRIDE + c] = v;
  }
  __syncthreads();

  int lane = threadIdx.x & 31;
  int wv   = threadIdx.x >> 5;       // N tile 0..7
  int half = lane >> 4;              // K sub-pair select (ISA A/B layout)
  int mr   = lane & 15;
  int ncol = wv * 16 + (lane & 15);

  v8f acc = {};
  for (int kb = 0; kb < INFEAT / 4; ++kb) {
    int k0 = kb * 4 + half * 2;      // lanes 0-15: K=0,1 ; lanes 16-31: K=2,3
    v2f a, b;
    a[0] = As[mr * AS_STRIDE + k0];
    a[1] = As[mr * AS_STRIDE + k0 + 1];
    b[0] = w_q[k0 * OUTF + ncol];
    b[1] = w_q[(k0 + 1) * OUTF + ncol];
    acc = __builtin_amdgcn_wmma_f32_16x16x4_f32(
        /*neg_a=*/false, a, /*neg_b=*/false, b,
        /*c_mod=*/(short)0, acc, /*reuse_a=*/false, /*reuse_b=*/false);
  }

  // epilogue: store feat tile + fused masked min/max for the next fq stage
  float mnH = INFINITY, mxH = -INFINITY, mnL = INFINITY, mxL = -INFINITY;
#pragma unroll
  for (int r = 0; r < 8; ++r) {
    int m = m0 + half * 8 + r;       // C layout: VGPR r -> M = r + 8*half
    float v = acc[r];
    feat[(long long)m * OUTF + ncol] = v;
    if (mask[m]) { mnH = fminf(mnH, v); mxH = fmaxf(mxH, v); }
    else         { mnL = fminf(mnL, v); mxL = fmaxf(mxL, v); }
  }
  commit_minmax4(slots, SL_F, mnH, mxH, mnL, mxL, s4);
}

// ---------------- K4: quantize feat in place ----------------
__global__ __launch_bounds__(256) void k_quant_feat(float* feat, const unsigned char* mask,
                                                    const unsigned* slots) {
  float mnH = finite0(fdec(slots[SL_F + 0])), mxH = finite0(fdec(slots[SL_F + 1]));
  float mnL = finite0(fdec(slots[SL_F + 2])), mxL = finite0(fdec(slots[SL_F + 3]));
  GRID_STRIDE(i, (long long)NNODES * OUTF) {
    float v = feat[i];
    feat[i] = mask[i >> 7] ? fq_apply(v, mnH, mxH, LVL_HI)
                           : fq_apply(v, mnL, mxL, LVL_LO);
  }
}

// ---------------- K5: edge attention logits (warp per edge) ----------------
__global__ __launch_bounds__(256) void k_alpha(const float* feat, const float* att_q,
                                               const int* src, const int* dst,
                                               const unsigned char* mask,
                                               float* alpha, unsigned* slots) {
  __shared__ unsigned s4[4];
  int lane = threadIdx.x & 31;
  long long warpId = blockIdx.x * (long long)(blockDim.x >> 5) + (threadIdx.x >> 5);
  long long nwarps = (long long)gridDim.x * (blockDim.x >> 5);
  float mnH = INFINITY, mxH = -INFINITY, mnL = INFINITY, mxL = -INFINITY;

  for (long long e = warpId; e < ETOT; e += nwarps) {
    int s = (e < NEDGES) ? src[e] : (int)(e - NEDGES);
    int d = (e < NEDGES) ? dst[e] : (int)(e - NEDGES);
    bool em = mask[s] != 0;
#pragma unroll
    for (int h = 0; h < HEADS; ++h) {
      float xi = feat[(long long)d * OUTF + h * CH + lane];
      float xj = feat[(long long)s * OUTF + h * CH + lane];
      float p = xi * att_q[h * 2 * CH + lane] + xj * att_q[h * 2 * CH + CH + lane];
#pragma unroll
      for (int o = 16; o > 0; o >>= 1) p += __shfl_xor(p, o, 32);
      if (lane == 0) {
        float a = (p > 0.0f) ? p : NSLOPE * p;   // leaky_relu
        alpha[e * HEADS + h] = a;
        if (em) { mnH = fminf(mnH, a); mxH = fmaxf(mxH, a); }
        else    { mnL = fminf(mnL, a); mxL = fmaxf(mxL, a); }
      }
    }
  }
  commit_minmax4(slots, SL_A, mnH, mxH, mnL, mxL, s4);
}

// ---------------- K6: quantize alpha + segment max per (dst, head) ---------
__global__ __launch_bounds__(256) void k_alpha_quant(float* alpha, const int* src,
                                                     const int* dst,
                                                     const unsigned char* mask,
                                                     const unsigned* slots,
                                                     unsigned* nodeMax) {
  float mnH = finite0(fdec(slots[SL_A + 0])), mxH = finite0(fdec(slots[SL_A + 1]));
  float mnL = finite0(fdec(slots[SL_A + 2])), mxL = finite0(fdec(slots[SL_A + 3]));
  GRID_STRIDE(i, (long long)ETOT * HEADS) {
    long long e = i >> 2; int h = (int)(i & 3);
    int s = (e < NEDGES) ? src[e] : (int)(e - NEDGES);
    int d = (e < NEDGES) ? dst[e] : (int)(e - NEDGES);
    float a = alpha[i];
    a = mask[s] ? fq_apply(a, mnH, mxH, LVL_HI) : fq_apply(a, mnL, mxL, LVL_LO);
    alpha[i] = a;
    atomicMax(&nodeMax[(long long)d * HEADS + h], fenc(a));
  }
}

// ---------------- K7: exp(alpha - segmax) + segment sum ----------------
__global__ __launch_bounds__(256) void k_alpha_exp(float* alpha, const int* dst,
                                                   const unsigned* nodeMax,
                                                   float* nodeSum) {
  GRID_STRIDE(i, (long long)ETOT * HEADS) {
    long long e = i >> 2; int h = (int)(i & 3);
    int d = (e < NEDGES) ? dst[e] : (int)(e - NEDGES);
    float mx = finite0(fdec(nodeMax[(long long)d * HEADS + h]));
    float ev = expf(alpha[i] - mx);
    alpha[i] = ev;
    atomicAdd(&nodeSum[(long long)d * HEADS + h], ev);
  }
}

// ---------------- K8: min/max of msg (recomputed, never stored) ------------
__global__ __launch_bounds__(256) void k_msg_minmax(const float* feat, const float* alpha,
                                                    const float* nodeSum, const int* src,
                                                    const int* dst, const unsigned char* mask,
                                                    unsigned* slots) {
  __shared__ unsigned s4[4];
  int lane = threadIdx.x & 31;
  long long warpId = blockIdx.x * (long long)(blockDim.x >> 5) + (threadIdx.x >> 5);
  long long nwarps = (long long)gridDim.x * (blockDim.x >> 5);
  float mnH = INFINITY, mxH = -INFINITY, mnL = INFINITY, mxL = -INFINITY;

  for (long long e = warpId; e < ETOT; e += nwarps) {
    int s = (e < NEDGES) ? src[e] : (int)(e - NEDGES);
    int d = (e < NEDGES) ? dst[e] : (int)(e - NEDGES);
    bool em = mask[s] != 0;
#pragma unroll
    for (int h = 0; h < HEADS; ++h) {
      float w = alpha[e * HEADS + h] / (nodeSum[(long long)d * HEADS + h] + 1e-16f);
      float v = feat[(long long)s * OUTF + h * CH + lane] * w;
      if (em) { mnH = fminf(mnH, v); mxH = fmaxf(mxH, v); }
      else    { mnL = fminf(mnL, v); mxL = fmaxf(mxL, v); }
    }
  }
  commit_minmax4(slots, SL_M, mnH, mxH, mnL, mxL, s4);
}

// ---------------- K9: quantized message scatter-add (segment sum) ----------
__global__ __launch_bounds__(256) void k_msg_scatter(const float* feat, const float* alpha,
                                                     const float* nodeSum, const int* src,
                                                     const int* dst, const unsigned char* mask,
                                                     const unsigned* slots, float* out) {
  float mnH = finite0(fdec(slots[SL_M + 0])), mxH = finite0(fdec(slots[SL_M + 1]));
  float mnL = finite0(fdec(slots[SL_M + 2])), mxL = finite0(fdec(slots[SL_M + 3]));
  int lane = threadIdx.x & 31;
  long long warpId = blockIdx.x * (long long)(blockDim.x >> 5) + (threadIdx.x >> 5);
  long long nwarps = (long long)gridDim.x * (blockDim.x >> 5);

  for (long long e = warpId; e < ETOT; e += nwarps) {
    int s = (e < NEDGES) ? src[e] : (int)(e - NEDGES);
    int d = (e < NEDGES) ? dst[e] : (int)(e - NEDGES);
    bool em = mask[s] != 0;
#pragma unroll
    for (int h = 0; h < HEADS; ++h) {
      float w = alpha[e * HEADS + h] / (nodeSum[(long long)d * HEADS + h] + 1e-16f);
      float v = feat[(long long)s * OUTF + h * CH + lane] * w;
      v = em ? fq_apply(v, mnH, mxH, LVL_HI) : fq_apply(v, mnL, mxL, LVL_LO);
      atomicAdd(&out[(long long)d * OUTF + h * CH + lane], v);
    }
  }
}

// ---------------- K10: min/max of aggr ----------------
__global__ __launch_bounds__(256) void k_minmax_aggr(const float* out, const unsigned char* mask,
                                                     unsigned* slots) {
  __shared__ unsigned s4[4];
  float mnH = INFINITY, mxH = -INFINITY, mnL = INFINITY, mxL = -INFINITY;
  GRID_STRIDE(i, (long long)NNODES * OUTF) {
    float v = out[i];
    if (mask[i >> 7]) { mnH = fminf(mnH, v); mxH = fmaxf(mxH, v); }
    else              { mnL = fminf(mnL, v); mxL = fmaxf(mxL, v); }
  }
  commit_minmax4(slots, SL_G, mnH, mxH, mnL, mxL, s4);
}

// ---------------- K11: aggr_q + bias, fused min/max of result --------------
__global__ __launch_bounds__(256) void k_out_bias(float* out, const float* bias,
                                                  const unsigned char* mask,
                                                  unsigned* slots) {
  __shared__ unsigned s4[4];
  float mnH = finite0(fdec(slots[SL_G + 0])), mxH = finite0(fdec(slots[SL_G + 1]));
  float mnL = finite0(fdec(slots[SL_G + 2])), mxL = finite0(fdec(slots[SL_G + 3]));
  float omnH = INFINITY, omxH = -INFINITY, omnL = INFINITY, omxL = -INFINITY;
  GRID_STRIDE(i, (long long)NNODES * OUTF) {
    float a = out[i];
    bool m = mask[i >> 7] != 0;
    a = m ? fq_apply(a, mnH, mxH, LVL_HI) : fq_apply(a, mnL, mxL, LVL_LO);
    float o = a + bias[i & 127];
    out[i] = o;
    if (m) { omnH = fminf(omnH, o); omxH = fmaxf(omxH, o); }
    else   { omnL = fminf(omnL, o); omxL = fmaxf(omxL, o); }
  }
  commit_minmax4(slots, SL_O, omnH, omxH, omnL, omxL, s4);
}

// ---------------- K12: final output quantization ----------------
__global__ __launch_bounds__(256) void k_out_quant(float* out, const unsigned char* mask,
                                                   const unsigned* slots) {
  float mnH = finite0(fdec(slots[SL_O + 0])), mxH = finite0(fdec(slots[SL_O + 1]));
  float mnL = finite0(fdec(slots[SL_O + 2])), mxL = finite0(fdec(slots[SL_O + 3]));
  GRID_STRIDE(i, (long long)NNODES * OUTF) {
    float v = out[i];
    out[i] = mask[i >> 7] ? fq_apply(v, mnH, mxH, LVL_HI)
                          : fq_apply(v, mnL, mxL, LVL_LO);
  }
}

// ---------------- launch ----------------
extern "C" void kernel_launch(void* const* d_in, const int* in_sizes, int n_in,
                              void* d_out, int out_size, void* d_ws, size_t ws_size,
                              hipStream_t stream) {
  const float* x    = (const float*)d_in[0];
  const float* w    = (const float*)d_in[1];
  const float* att  = (const float*)d_in[2];
  const float* bias = (const float*)d_in[3];
  const int*   ei   = (const int*)d_in[4];
  const int*   src  = ei;            // edge_index[0]
  const int*   dst  = ei + NEDGES;   // edge_index[1]
  const unsigned char* mask = (const unsigned char*)d_in[5];  // jnp bool = 1 byte
  float* out = (float*)d_out;

  // workspace carve-up (~41 MB)
  char* p = (char*)d_ws;
  size_t off = 0;
  unsigned* slots   = (unsigned*)(p + off); off += 256;
  float*    w_q     = (float*)(p + off);    off += (size_t)INFEAT * OUTF * 4;       // 64 KB
  float*    att_q   = (float*)(p + off);    off += (size_t)HEADS * 2 * CH * 4;      // 1 KB
  unsigned* nodeMax = (unsigned*)(p + off); off += (size_t)NNODES * HEADS * 4;      // 0.8 MB
  float*    nodeSum = (float*)(p + off);    off += (size_t)NNODES * HEADS * 4;      // 0.8 MB
  float*    feat    = (float*)(p + off);    off += (size_t)NNODES * OUTF * 4;       // 25.6 MB
  float*    alpha   = (float*)(p + off);    off += (size_t)ETOT * HEADS * 4;        // 13.6 MB
  (void)ws_size; (void)in_sizes; (void)n_in; (void)out_size;

  const int T = 256;
  const int GBIG  = 4096;   // grid-stride elementwise/reduction kernels
  const int GWARP = 8192;   // warp-per-edge kernels (8 warps/block)

  k_init       <<<GBIG, T, 0, stream>>>(out, slots, nodeMax, nodeSum);
  k_minmax_x   <<<GBIG, T, 0, stream>>>(x, mask, slots);
  k_minmax_watt<<<1,    T, 0, stream>>>(w, att, slots);
  k_quant_watt <<<64,   T, 0, stream>>>(w, att, slots, w_q, att_q);
  k_gemm_feat  <<<NNODES / 16, T, 0, stream>>>(x, mask, slots, w_q, feat);
  k_quant_feat <<<GBIG, T, 0, stream>>>(feat, mask, slots);
  k_alpha      <<<GWARP, T, 0, stream>>>(feat, att_q, src, dst, mask, alpha, slots);
  k_alpha_quant<<<GBIG, T, 0, stream>>>(alpha, src, dst, mask, slots, nodeMax);
  k_alpha_exp  <<<GBIG, T, 0, stream>>>(alpha, dst, nodeMax, nodeSum);
  k_msg_minmax <<<GWARP, T, 0, stream>>>(feat, alpha, nodeSum, src, dst, mask, slots);
  k_msg_scatter<<<GWARP, T, 0, stream>>>(feat, alpha, nodeSum, src, dst, mask, slots, out);
  k_minmax_aggr<<<GBIG, T, 0, stream>>>(out, mask, slots);
  k_out_bias   <<<GBIG, T, 0, stream>>>(out, bias, mask, slots);
  k_out_quant  <<<GBIG, T, 0, stream>>>(out, mask, slots);
}